// GATLayer_14448269984592
// MI455X (gfx1250) — compile-verified
//
#include <hip/hip_runtime.h>
#include <math.h>

// ---------------------------------------------------------------------------
// Types for CDNA5 WMMA f32 16x16x4:  A 16x4 (2 VGPRs), B 4x16 (2 VGPRs),
// C/D 16x16 (8 VGPRs).  D = A*B + C, full fp32 — matches reference precision.
// ---------------------------------------------------------------------------
typedef float v2f __attribute__((ext_vector_type(2)));
typedef float v8f __attribute__((ext_vector_type(8)));

__device__ __forceinline__ v8f wmma4(v2f a, v2f b, v8f c) {
    return __builtin_amdgcn_wmma_f32_16x16x4_f32(
        /*neg_a=*/false, a, /*neg_b=*/false, b,
        /*c_mod=*/(short)0, c, /*reuse_a=*/false, /*reuse_b=*/false);
}

// float atomic max via sign-aware int/uint atomics (works on LDS and global)
__device__ __forceinline__ void atomicMaxF(float* addr, float val) {
    if (val >= 0.0f)
        atomicMax((int*)addr, __float_as_int(val));
    else
        atomicMin((unsigned int*)addr, __float_as_uint(val));
}

#define NHh 4
#define NRr 4
#define Ff  64

// ---------------------------------------------------------------------------
// Generic WMMA f32 GEMM:  C[M,NC] = A[M,K] @ W[NC,K]^T
// One wave computes a 16x64 tile (4 accumulators, A fragment reused 4x).
// K multiple of 4; M multiple of 16; NC multiple of 64.
// ---------------------------------------------------------------------------
__global__ __launch_bounds__(256) void gemm_wmma_f32(
    const float* __restrict__ A, const float* __restrict__ W,
    float* __restrict__ C, int M16, int tilesN, int K, int NC)
{
    int wid = (blockIdx.x * blockDim.x + threadIdx.x) >> 5;
    if (wid >= M16 * tilesN) return;
    int tm = wid / tilesN;
    int tn = wid - tm * tilesN;
    int lane = threadIdx.x & 31;
    int lmod = lane & 15;
    int ksel = (lane >> 4) << 1;       // K offset 0 or 2 for this half-wave
    int rowhalf = (lane >> 4) << 3;    // C rows: +0 or +8

    const float* Ar = A + (size_t)(tm * 16 + lmod) * K + ksel;
    const float* Wr = W + (size_t)(tn * 64 + lmod) * K + ksel;
    const int wstep = 16 * K;

    v8f acc0 = {}, acc1 = {}, acc2 = {}, acc3 = {};
    for (int k = 0; k < K; k += 4) {
        v2f a  = { Ar[k],             Ar[k + 1] };
        v2f b0 = { Wr[k],             Wr[k + 1] };
        v2f b1 = { Wr[wstep + k],     Wr[wstep + k + 1] };
        v2f b2 = { Wr[2 * wstep + k], Wr[2 * wstep + k + 1] };
        v2f b3 = { Wr[3 * wstep + k], Wr[3 * wstep + k + 1] };
        acc0 = wmma4(a, b0, acc0);
        acc1 = wmma4(a, b1, acc1);
        acc2 = wmma4(a, b2, acc2);
        acc3 = wmma4(a, b3, acc3);
    }
#pragma unroll
    for (int v = 0; v < 8; ++v) {
        size_t row = (size_t)(tm * 16 + v + rowhalf);
        float* Cr = C + row * NC + tn * 64 + lmod;
        Cr[0]  = acc0[v];
        Cr[16] = acc1[v];
        Cr[32] = acc2[v];
        Cr[48] = acc3[v];
    }
}

// ---------------------------------------------------------------------------
// s_src/s_trg dots: one thread per (n,h,r), reads proj row once for both.
// layouts: proj (N,NH,NR,F); s (N,NH,NR)
// ---------------------------------------------------------------------------
__global__ __launch_bounds__(256) void score_dots(
    const float* __restrict__ proj,
    const float* __restrict__ score_src, const float* __restrict__ score_trg,
    float* __restrict__ s_src, float* __restrict__ s_trg, int total)
{
    int i = blockIdx.x * blockDim.x + threadIdx.x;
    if (i >= total) return;
    int hr = i & 15;                           // (h*NR + r)
    const float* p = proj + (size_t)i * Ff;
    const float* a = score_src + hr * Ff;
    const float* b = score_trg + hr * Ff;
    float ss = 0.f, st = 0.f;
#pragma unroll 8
    for (int f = 0; f < Ff; ++f) { float pv = p[f]; ss += pv * a[f]; st += pv * b[f]; }
    s_src[i] = ss;
    s_trg[i] = st;
}

// ---------------------------------------------------------------------------
// Edge scores + leaky relu + per-(rel,head) max (block-local LDS reduction
// first, then 16 global atomics per block).
// ---------------------------------------------------------------------------
__global__ __launch_bounds__(256) void edge_scores(
    const float* __restrict__ s_src, const float* __restrict__ s_trg,
    const int* __restrict__ src, const int* __restrict__ trg,
    const int* __restrict__ rel,
    float* __restrict__ es, float* __restrict__ m, int E)
{
    __shared__ float lmax[16];
    if (threadIdx.x < 16) lmax[threadIdx.x] = -INFINITY;
    __syncthreads();
    int e = blockIdx.x * blockDim.x + threadIdx.x;
    if (e < E) {
        int s = src[e], t = trg[e], r = rel[e];
#pragma unroll
        for (int h = 0; h < NHh; ++h) {
            float v = s_src[((size_t)s * NHh + h) * NRr + r]
                    + s_trg[((size_t)t * NHh + h) * NRr + r];
            v = (v > 0.f) ? v : 0.2f * v;          // leaky_relu(0.2)
            es[(size_t)e * NHh + h] = v;
            atomicMaxF(&lmax[r * NHh + h], v);
        }
    }
    __syncthreads();
    if (threadIdx.x < 16) atomicMaxF(&m[threadIdx.x], lmax[threadIdx.x]);
}

// exp(e - m[rel]) + segment-sum denominators
__global__ __launch_bounds__(256) void edge_exp(
    const int* __restrict__ trg, const int* __restrict__ rel,
    const float* __restrict__ m, float* __restrict__ es,
    float* __restrict__ denom, int total)
{
    int i = blockIdx.x * blockDim.x + threadIdx.x;
    if (i >= total) return;
    int e = i >> 2, h = i & 3;
    int r = rel[e], t = trg[e];
    float x = expf(es[i] - m[r * NHh + h]);
    es[i] = x;                                           // reuse buffer as e_exp
    unsafeAtomicAdd(&denom[((size_t)t * NRr + r) * NHh + h], x);
}

// att = e_exp / (denom[seg] + 1e-16)
__global__ __launch_bounds__(256) void edge_att(
    const int* __restrict__ trg, const int* __restrict__ rel,
    const float* __restrict__ denom, float* __restrict__ es, int total)
{
    int i = blockIdx.x * blockDim.x + threadIdx.x;
    if (i >= total) return;
    int e = i >> 2, h = i & 3;
    size_t seg = (size_t)trg[e] * NRr + rel[e];
    es[i] = es[i] / (denom[seg * NHh + h] + 1e-16f);     // in-place -> att
}

// Per-edge weighted scatter: one block per edge (256 threads = NH*F lanes,
// consecutive lanes -> consecutive agg addresses: coalesced HW fp32 atomics)
// agg layout (N*NR, NH, F) i.e. (n, r, h, f)
__global__ __launch_bounds__(256) void scatter_edges(
    const float* __restrict__ proj, const float* __restrict__ att,
    const int* __restrict__ src, const int* __restrict__ trg,
    const int* __restrict__ rel, float* __restrict__ agg)
{
    int e = blockIdx.x;
    int h = threadIdx.x >> 6, f = threadIdx.x & 63;
    int s = src[e], t = trg[e], r = rel[e];
    float a = att[(size_t)e * NHh + h];
    float v = proj[(((size_t)s * NHh + h) * NRr + r) * Ff + f] * a;
    unsafeAtomicAdd(&agg[(((size_t)t * NRr + r) * NHh + h) * Ff + f], v);
}

// ---------------------------------------------------------------------------
// Fused MLP: rows = agg viewed as (N*NR*NH, 64).
//   h1 = relu(rows@W1^T + b1)   (WMMA, kept in regs -> LDS)
//   h2 = relu(h1@W2^T + b2)     (WMMA from LDS)
//   sc = mish(h2@W3^T + b3)     (per-row dot, 16 lanes)
// Never spills h1/h2 to HBM (saves ~800 MB of traffic).
// One wave = 16 rows x full 64 cols; 8 waves/block; 32 KB LDS.
// ---------------------------------------------------------------------------
__global__ __launch_bounds__(256) void mlp_fused(
    const float* __restrict__ agg,
    const float* __restrict__ W1, const float* __restrict__ b1,
    const float* __restrict__ W2, const float* __restrict__ b2,
    const float* __restrict__ W3, const float* __restrict__ b3,
    float* __restrict__ sc, int rowTiles)
{
    __shared__ float lds[8 * 16 * 64];
    int wid = (blockIdx.x * blockDim.x + threadIdx.x) >> 5;
    if (wid >= rowTiles) return;
    int lane = threadIdx.x & 31;
    int lmod = lane & 15;
    int ksel = (lane >> 4) << 1;
    int rowhalf = (lane >> 4) << 3;
    float* L = &lds[(threadIdx.x >> 5) * 16 * 64];
    size_t rowBase = (size_t)wid * 16;

    // ---- layer 1: WMMA from global agg ----
    const float* Ar  = agg + (rowBase + lmod) * Ff + ksel;
    const float* W1r = W1 + lmod * 64 + ksel;
    v8f acc[4] = {};
    for (int k = 0; k < 64; k += 4) {
        v2f a = { Ar[k], Ar[k + 1] };
#pragma unroll
        for (int t = 0; t < 4; ++t) {
            v2f b = { W1r[t * 1024 + k], W1r[t * 1024 + k + 1] };
            acc[t] = wmma4(a, b, acc[t]);
        }
    }
#pragma unroll
    for (int t = 0; t < 4; ++t)
#pragma unroll
        for (int v = 0; v < 8; ++v) {
            int col = t * 16 + lmod;
            L[(v + rowhalf) * 64 + col] = fmaxf(acc[t][v] + b1[col], 0.f);
        }

    // ---- layer 2: WMMA from LDS (same-wave DS ops are in order) ----
    v8f acc2[4] = {};
    const float* W2r = W2 + lmod * 64 + ksel;
    const float* Lr  = L + lmod * 64 + ksel;
    for (int k = 0; k < 64; k += 4) {
        v2f a = { Lr[k], Lr[k + 1] };
#pragma unroll
        for (int t = 0; t < 4; ++t) {
            v2f b = { W2r[t * 1024 + k], W2r[t * 1024 + k + 1] };
            acc2[t] = wmma4(a, b, acc2[t]);
        }
    }
#pragma unroll
    for (int t = 0; t < 4; ++t)
#pragma unroll
        for (int v = 0; v < 8; ++v) {
            int col = t * 16 + lmod;
            L[(v + rowhalf) * 64 + col] = fmaxf(acc2[t][v] + b2[col], 0.f);
        }

    // ---- score head: sc = mish(h2 . W3 + b3), lanes 0..15 own one row each
    if (lane < 16) {
        float s = b3[0];
        const float* hr = L + lane * 64;
#pragma unroll 8
        for (int f = 0; f < 64; ++f) s += hr[f] * W3[f];
        float sp = (s > 20.f) ? s : log1pf(expf(s));     // softplus
        sc[rowBase + lane] = s * tanhf(sp);              // mish
    }
}

// ---------------------------------------------------------------------------
// Finalize per node: relation softmax of sc, weighted sum of agg over r,
// + skip + bias, ELU, LayerNorm over NH*F=256.  One block per node.
// ---------------------------------------------------------------------------
__global__ __launch_bounds__(256) void finalize(
    const float* __restrict__ agg, const float* __restrict__ sc,
    const float* __restrict__ skipb, const float* __restrict__ bias,
    const float* __restrict__ gamma, const float* __restrict__ beta,
    float* __restrict__ out)
{
    __shared__ float scl[16];     // [h][r]
    __shared__ float a_rel[16];   // [h][r]
    __shared__ float red[256];
    __shared__ float stat[2];
    int n = blockIdx.x;
    int t = threadIdx.x;

    if (t < 16) {
        int h = t & 3, r = t >> 2;
        scl[h * NRr + r] = sc[((size_t)n * NRr + r) * NHh + h];
    }
    __syncthreads();
    if (t < 4) {                   // softmax over r for head h=t
        float mx = scl[t * 4];
        for (int r = 1; r < 4; ++r) mx = fmaxf(mx, scl[t * 4 + r]);
        float ex[4], s = 0.f;
        for (int r = 0; r < 4; ++r) { ex[r] = expf(scl[t * 4 + r] - mx); s += ex[r]; }
        for (int r = 0; r < 4; ++r) a_rel[t * 4 + r] = ex[r] / s;
    }
    __syncthreads();

    int h = t >> 6, f = t & 63;
    float v = 0.f;
#pragma unroll
    for (int r = 0; r < 4; ++r)
        v += agg[(((size_t)n * NRr + r) * NHh + h) * Ff + f] * a_rel[h * 4 + r];
    v += skipb[(size_t)n * 256 + t];          // projected skip (n, h*64+f)
    v += bias[t];
    v = (v > 0.f) ? v : expm1f(v);            // ELU

    red[t] = v;
    __syncthreads();
    for (int s = 128; s > 0; s >>= 1) { if (t < s) red[t] += red[t + s]; __syncthreads(); }
    if (t == 0) stat[0] = red[0] * (1.f / 256.f);
    __syncthreads();
    float d = v - stat[0];
    red[t] = d * d;
    __syncthreads();
    for (int s = 128; s > 0; s >>= 1) { if (t < s) red[t] += red[t + s]; __syncthreads(); }
    if (t == 0) stat[1] = red[0] * (1.f / 256.f);
    __syncthreads();
    out[(size_t)n * 256 + t] = d * rsqrtf(stat[1] + 1e-5f) * gamma[t] + beta[t];
}

__global__ void init_m(float* m) {
    if (threadIdx.x < 16) m[threadIdx.x] = -INFINITY;
}

// ---------------------------------------------------------------------------
extern "C" void kernel_launch(void* const* d_in, const int* in_sizes, int n_in,
                              void* d_out, int out_size, void* d_ws, size_t ws_size,
                              hipStream_t stream)
{
    const float* x         = (const float*)d_in[0];
    const int*   src       = (const int*)  d_in[1];
    const int*   trg       = (const int*)  d_in[2];
    const int*   rel       = (const int*)  d_in[3];
    const float* W_proj    = (const float*)d_in[5];
    const float* score_src = (const float*)d_in[6];
    const float* score_trg = (const float*)d_in[7];
    const float* W1        = (const float*)d_in[8];
    const float* b1        = (const float*)d_in[9];
    const float* W2        = (const float*)d_in[10];
    const float* b2        = (const float*)d_in[11];
    const float* W3        = (const float*)d_in[12];
    const float* b3        = (const float*)d_in[13];
    const float* W_skip    = (const float*)d_in[14];
    const float* bias      = (const float*)d_in[15];
    const float* gamma     = (const float*)d_in[16];
    const float* beta      = (const float*)d_in[17];
    float* out = (float*)d_out;

    const int N = in_sizes[0] / 256;   // FIN=256
    const int E = in_sizes[1];

    // workspace layout (floats)
    float* ws = (float*)d_ws;
    size_t o = 0;
    float* proj  = ws + o; o += (size_t)N * 1024;   // (N,NH,NR,F)
    float* agg   = ws + o; o += (size_t)N * 1024;   // (N,NR,NH,F)
    float* skipb = ws + o; o += (size_t)N * 256;    // (N,NH*F)
    float* s_src = ws + o; o += (size_t)N * 16;
    float* s_trg = ws + o; o += (size_t)N * 16;
    float* es    = ws + o; o += (size_t)E * 4;      // e_s -> e_exp -> att
    float* denom = ws + o; o += (size_t)N * 16;
    float* scbuf = ws + o; o += (size_t)N * 16;
    float* mbuf  = ws + o; o += 16;

    // per-launch (re)initialization of accumulated buffers
    init_m<<<1, 32, 0, stream>>>(mbuf);
    hipMemsetAsync(denom, 0, sizeof(float) * (size_t)N * 16, stream);
    hipMemsetAsync(agg,   0, sizeof(float) * (size_t)N * 1024, stream);

    const int M16 = N / 16;   // 3125

    // proj = x @ W_proj^T   (N,1024)
    gemm_wmma_f32<<<(M16 * 16 + 7) / 8, 256, 0, stream>>>(x, W_proj, proj, M16, 16, 256, 1024);
    // per-node attention score dots
    score_dots<<<(N * 16 + 255) / 256, 256, 0, stream>>>(proj, score_src, score_trg, s_src, s_trg, N * 16);
    // skip = x @ W_skip^T   (N,256)
    gemm_wmma_f32<<<(M16 * 4 + 7) / 8, 256, 0, stream>>>(x, W_skip, skipb, M16, 4, 256, 256);
    // edge attention pipeline
    edge_scores<<<(E + 255) / 256, 256, 0, stream>>>(s_src, s_trg, src, trg, rel, es, mbuf, E);
    edge_exp<<<(E * 4 + 255) / 256, 256, 0, stream>>>(trg, rel, mbuf, es, denom, E * 4);
    edge_att<<<(E * 4 + 255) / 256, 256, 0, stream>>>(trg, rel, denom, es, E * 4);
    scatter_edges<<<E, 256, 0, stream>>>(proj, es, src, trg, rel, agg);
    // fused relation-MLP head -> sc   (rows = N*NR*NH, tiles of 16)
    mlp_fused<<<(N + 7) / 8, 256, 0, stream>>>(agg, W1, b1, W2, b2, W3, b3, scbuf, N);
    // softmax over relations, aggregate, skip, ELU, LayerNorm
    finalize<<<N, 256, 0, stream>>>(agg, scbuf, skipb, bias, gamma, beta, out);
}